// PSFmap_Malleconv_47854525612208
// MI455X (gfx1250) — compile-verified
//
#include <hip/hip_runtime.h>
#include <hip/hip_bf16.h>

typedef __attribute__((ext_vector_type(2))) float v2f;
typedef __attribute__((ext_vector_type(8))) float v8f;

#define N_BATCH 4
#define C_IN    32
#define H_IMG   256
#define W_IMG   256
#define C_OUT   288           // C*9
#define HW_SM   1024          // 32*32
#define M_GEMM  4096          // N_BATCH * 32 * 32 pixels
#define K_GEMM  32
#define N_GEMM  288

// ---------------------------------------------------------------------------
// Kernel 1: psf [4,32,256,256] -> 1/4 bilinear (degenerates to fixed 2x2 avg,
// weights are exactly 0.5) -> 2x2 maxpool  =>  p [4,32,32,32]
// ---------------------------------------------------------------------------
__global__ void k_downsample(const float* __restrict__ psf, float* __restrict__ p) {
    int idx = blockIdx.x * blockDim.x + threadIdx.x;   // 0 .. 131071
    int j  = idx & 31;
    int i  = (idx >> 5) & 31;
    int nc = idx >> 10;                                // 0 .. 127
    const float* src = psf + (size_t)nc * (H_IMG * W_IMG);

    float m = -3.402823466e38f;
    #pragma unroll
    for (int di = 0; di < 2; ++di) {
        #pragma unroll
        for (int dj = 0; dj < 2; ++dj) {
            int a = 2 * i + di, b = 2 * j + dj;        // 64x64 intermediate coords
            int r0 = 4 * a + 1, r1 = 4 * a + 2;        // src=(a+0.5)*4-0.5 = 4a+1.5
            int c0 = 4 * b + 1, c1 = 4 * b + 2;
            float v = 0.25f * (src[r0 * W_IMG + c0] + src[r0 * W_IMG + c1] +
                               src[r1 * W_IMG + c0] + src[r1 * W_IMG + c1]);
            m = fmaxf(m, v);
        }
    }
    p[idx] = m;
}

// ---------------------------------------------------------------------------
// Kernel 2: KPResBlock as dual WMMA f32 GEMM.
//   Y[m][o] = lrelu(P[m][:] . w1[o][:] + b1[o]) + (P[m][:] . ws[o][:] + bs[o])
//   M=4096 (m = n*1024 + h*32 + w), K=32, N=288.
// One wave computes a 16x16 tile via 8 x V_WMMA_F32_16X16X4_F32 per chain.
// Grid exactly covers all tiles -> EXEC all ones for every WMMA (ISA req).
// ---------------------------------------------------------------------------
__global__ void k_kpres_gemm(const float* __restrict__ p,
                             const float* __restrict__ w1, const float* __restrict__ b1,
                             const float* __restrict__ wsm, const float* __restrict__ bs,
                             float* __restrict__ y) {
    int wave = (blockIdx.x * blockDim.x + threadIdx.x) >> 5;   // 0 .. 4607
    int lane = threadIdx.x & 31;
    const int MT = M_GEMM / 16;                                // 256
    int mt = wave % MT;
    int nt = wave / MT;                                        // 0 .. 17
    int m_base = mt * 16;
    int o_base = nt * 16;

    int laneHalf = lane >> 4;        // 0: K pair {k0,k0+1}, 1: {k0+2,k0+3}
    int laneMod  = lane & 15;

    // A operand: row m = m_base + laneMod, p laid out [N,C,32,32]
    int m  = m_base + laneMod;
    int n  = m >> 10;
    int hw = m & 1023;
    const float* pa = p + (size_t)n * (C_IN * HW_SM) + hw;     // + c*1024

    // B operand: column o = o_base + laneMod, weights laid out [288,32]
    const float* w1row = w1  + (size_t)(o_base + laneMod) * K_GEMM;
    const float* wsrow = wsm + (size_t)(o_base + laneMod) * K_GEMM;

    v8f acc1 = {};
    v8f acc2 = {};
    #pragma unroll
    for (int k0 = 0; k0 < K_GEMM; k0 += 4) {
        int ka = k0 + laneHalf * 2;
        v2f a;  a.x  = pa[(size_t)ka * HW_SM];  a.y  = pa[(size_t)(ka + 1) * HW_SM];
        v2f bA; bA.x = w1row[ka];               bA.y = w1row[ka + 1];
        v2f bB; bB.x = wsrow[ka];               bB.y = wsrow[ka + 1];
        acc1 = __builtin_amdgcn_wmma_f32_16x16x4_f32(false, a, false, bA,
                                                     (short)0, acc1, false, false);
        acc2 = __builtin_amdgcn_wmma_f32_16x16x4_f32(false, a, false, bB,
                                                     (short)0, acc2, false, false);
    }

    // D layout: VGPR r -> (M = r + 8*laneHalf, N = laneMod)
    #pragma unroll
    for (int r = 0; r < 8; ++r) {
        int mm = m_base + r + laneHalf * 8;
        int oo = o_base + laneMod;
        int nn  = mm >> 10;
        int hw2 = mm & 1023;
        float t1 = acc1[r] + b1[oo];
        float t2 = acc2[r] + bs[oo];
        float v  = (t1 >= 0.0f ? t1 : 0.2f * t1) + t2;
        y[((size_t)nn * C_OUT + oo) * HW_SM + hw2] = v;
    }
}

// ---------------------------------------------------------------------------
// Kernel 3: fused x8 bilinear upsample of y [4,288,32,32] + per-pixel dynamic
// 3x3 filter over featuremap (zero-padded). Never materializes the 302 MB
// upsampled filter tensor; y (4.7 MB) is served from L2.
// Block = 256 threads = one 16x16 spatial tile of one (n,c).
// ---------------------------------------------------------------------------
__global__ void k_fac(const float* __restrict__ fm, const float* __restrict__ y,
                      float* __restrict__ out) {
    int t  = threadIdx.x;
    int tx = t & 15, ty = t >> 4;
    int b  = blockIdx.x;
    int tilex = b & 15;
    int tiley = (b >> 4) & 15;
    int nc    = b >> 8;                    // 0..127  (n*32 + c)
    int h = tiley * 16 + ty;
    int w = tilex * 16 + tx;

    // align_corners=False, scale 256 -> 32
    float sh = fmaxf((h + 0.5f) * 0.125f - 0.5f, 0.0f);
    float sw = fmaxf((w + 0.5f) * 0.125f - 0.5f, 0.0f);
    int y0 = (int)sh; int y1 = y0 + 1; if (y1 > 31) y1 = 31;
    int x0 = (int)sw; int x1 = x0 + 1; if (x1 > 31) x1 = 31;
    float wy = sh - (float)y0;
    float wx = sw - (float)x0;

    int n = nc >> 5, c = nc & 31;
    const float* yb = y + ((size_t)n * C_OUT + (size_t)c * 9) * HW_SM;

    float filt[9];
    #pragma unroll
    for (int k = 0; k < 9; ++k) {
        const float* yk = yb + (size_t)k * HW_SM;
        float v00 = yk[y0 * 32 + x0];
        float v01 = yk[y0 * 32 + x1];
        float v10 = yk[y1 * 32 + x0];
        float v11 = yk[y1 * 32 + x1];
        float top = v00 + wx * (v01 - v00);
        float bot = v10 + wx * (v11 - v10);
        filt[k] = top + wy * (bot - top);
    }

    const float* f = fm + (size_t)nc * (H_IMG * W_IMG);
    float acc = 0.0f;
    #pragma unroll
    for (int dy = 0; dy < 3; ++dy) {
        int hh = h + dy - 1;
        #pragma unroll
        for (int dx = 0; dx < 3; ++dx) {
            int ww = w + dx - 1;
            float v = (hh >= 0 && hh < H_IMG && ww >= 0 && ww < W_IMG)
                          ? f[hh * W_IMG + ww] : 0.0f;
            acc = fmaf(v, filt[dy * 3 + dx], acc);
        }
    }
    out[(size_t)nc * (H_IMG * W_IMG) + h * W_IMG + w] = acc;
}

// ---------------------------------------------------------------------------
extern "C" void kernel_launch(void* const* d_in, const int* in_sizes, int n_in,
                              void* d_out, int out_size, void* d_ws, size_t ws_size,
                              hipStream_t stream) {
    const float* psf = (const float*)d_in[0];
    const float* fm  = (const float*)d_in[1];
    const float* w1  = (const float*)d_in[2];
    const float* b1  = (const float*)d_in[3];
    const float* wsm = (const float*)d_in[4];
    const float* bs  = (const float*)d_in[5];
    float* out = (float*)d_out;

    float* p = (float*)d_ws;                       // 4*32*32*32   = 131072 f32 (512 KB)
    float* y = p + (N_BATCH * C_IN * HW_SM);       // 4*288*32*32  = 1179648 f32 (4.7 MB)

    // Stage 1: 131072 elements
    k_downsample<<<(N_BATCH * C_IN * HW_SM) / 256, 256, 0, stream>>>(psf, p);

    // Stage 2: 4608 wave-tiles * 32 lanes = 147456 threads -> 576 blocks of 256
    k_kpres_gemm<<<576, 256, 0, stream>>>(p, w1, b1, wsm, bs, y);

    // Stage 3: 128 (n,c) * 16 * 16 tiles = 32768 blocks
    k_fac<<<32768, 256, 0, stream>>>(fm, y, out);
}